// ComplexUpSampling2D_79594333930123
// MI455X (gfx1250) — compile-verified
//
#include <hip/hip_runtime.h>

// ComplexUpSampling2D: nearest-neighbor 2x upsample of a complex tensor.
// x_real, x_imag: [8,128,128,64] f32  ->  out: [8,256,256,64] complex64
// (interleaved re/im float pairs). Pure data movement: ~256 MiB stores +
// 64 MiB reads => HBM-bound (~13.5 us floor @ 23.3 TB/s). Strategy:
//  - one block per output row; stage the mapped input row (64 KB) in LDS
//    via CDNA5 async global->LDS b128 copies (ASYNCcnt path),
//  - emit the output row as contiguous non-temporal b128 stores so the
//    streamed 256 MiB output doesn't evict the L2-resident inputs.

typedef float f2 __attribute__((ext_vector_type(2)));
typedef float f4 __attribute__((ext_vector_type(4)));
typedef int   v4i __attribute__((vector_size(16)));  // matches builtin param type

#if defined(__has_builtin)
#  if __has_builtin(__builtin_amdgcn_global_load_async_to_lds_b128)
#    define USE_ASYNC_LDS 1
#  endif
#endif
#ifndef USE_ASYNC_LDS
#  define USE_ASYNC_LDS 0
#endif

#define B_  8
#define H_  128
#define W_  128
#define C_  64
#define HO_ 256
#define WO_ 256

// rows = clip(round_half_even(H * i / H_out)) with H_out = 2H:
// value is i/2 exactly; for odd i it is k+0.5 (k = i>>1) which rounds to the
// even neighbor => idx = k + (k&1). Closed form (works for even i too):
__device__ __forceinline__ int nn_map(int i) {
    int k = i >> 1;
    int idx = k + (i & k & 1);
    return idx > (H_ - 1) ? (H_ - 1) : idx;
}

__global__ __launch_bounds__(256) void ComplexUpSampling2D_79594333930123_kernel(
    const float* __restrict__ xr, const float* __restrict__ xi,
    float* __restrict__ out) {
    // 64 KB LDS: [0,2048) f4 = real input row, [2048,4096) f4 = imag input row
    __shared__ f4 smem[4096];

    const int tid = threadIdx.x;
    const int bid = blockIdx.x;         // 0 .. B_*HO_-1
    const int b   = bid >> 8;           // batch
    const int ho  = bid & (HO_ - 1);    // output row
    const int ri  = nn_map(ho);         // mapped input row

    const size_t in_row = (((size_t)b * H_ + ri) * W_) * C_;  // floats
    const f4* gre = (const f4*)(xr + in_row);  // 2048 x f4 (32 KB)
    const f4* gim = (const f4*)(xi + in_row);

#if USE_ASYNC_LDS
    // CDNA5 async global->LDS DMA path (tracked by ASYNCcnt).
#pragma unroll
    for (int k = 0; k < 8; ++k) {
        const int j = tid + (k << 8);
        __builtin_amdgcn_global_load_async_to_lds_b128(
            (v4i*)(gre + j), (v4i*)(&smem[j]), 0, 0);
        __builtin_amdgcn_global_load_async_to_lds_b128(
            (v4i*)(gim + j), (v4i*)(&smem[2048 + j]), 0, 0);
    }
    asm volatile("s_wait_asynccnt 0" ::: "memory");
    __syncthreads();
#else
#pragma unroll
    for (int k = 0; k < 8; ++k) {
        const int j = tid + (k << 8);
        smem[j]        = gre[j];
        smem[2048 + j] = gim[j];
    }
    __syncthreads();
#endif

    const float* s_re = (const float*)smem;
    const float* s_im = (const float*)(smem + 2048);
    float* orow = out + (((size_t)b * HO_ + ho) * WO_) * (size_t)C_ * 2;

    // 256 output pixels * 32 lane-chunks (2 channels each) = 8192 chunks;
    // 32 iterations of 256 threads. Each wave handles exactly one pixel per
    // iteration: conflict-free LDS float2 reads, contiguous 4 KB NT stores.
#pragma unroll 4
    for (int it = 0; it < 32; ++it) {
        const int gid = (it << 8) + tid;
        const int wo  = gid >> 5;          // output column
        const int cp  = (gid & 31) << 1;   // channel pair start
        const int ci  = nn_map(wo);        // mapped input column
        const f2 re = *(const f2*)(s_re + ci * C_ + cp);
        const f2 im = *(const f2*)(s_im + ci * C_ + cp);
        f4 o;
        o.x = re.x; o.y = im.x; o.z = re.y; o.w = im.y;
        __builtin_nontemporal_store(
            o, (f4*)(orow + ((size_t)(wo * C_ + cp) << 1)));
    }
}

extern "C" void kernel_launch(void* const* d_in, const int* in_sizes, int n_in,
                              void* d_out, int out_size, void* d_ws, size_t ws_size,
                              hipStream_t stream) {
    (void)in_sizes; (void)n_in; (void)out_size; (void)d_ws; (void)ws_size;
    const float* xr = (const float*)d_in[0];
    const float* xi = (const float*)d_in[1];
    float* out = (float*)d_out;
    ComplexUpSampling2D_79594333930123_kernel<<<B_ * HO_, 256, 0, stream>>>(xr, xi, out);
}